// lgn_frame_53506702573593
// MI455X (gfx1250) — compile-verified
//
#include <hip/hip_runtime.h>

// ---------------- problem constants (match reference) ----------------
#define N_USERS   100000
#define N_ITEMS   50000
#define N_NODES   150000          // N_USERS + N_ITEMS
#define DIM       64
#define NHOPS     3
#define ROWSTRIDE 256             // (NHOPS+1) * DIM floats per node in d_out
#define CHUNK     32              // edges per wave chunk
#define WPB       8               // waves per block (256 threads, wave32)

// d_out layout: out[n*256 + hop*64 + d]; hop slice h is the post-dropout agg
// that feeds hop h+1, so no scratch workspace is needed at all.

// ---------------- init: hop0 = concat(user,item); zero hops 1..3 ----------------
__global__ void lgn_init_kernel(const float* __restrict__ ue,
                                const float* __restrict__ ie,
                                float* __restrict__ out) {
    int idx = blockIdx.x * blockDim.x + threadIdx.x;   // one float4 of hop0
    const int total = N_NODES * (DIM / 4);
    if (idx >= total) return;
    int n = idx >> 4;          // node
    int q = idx & 15;          // float4 slot within 64 dims
    float4 v;
    if (n < N_USERS) v = ((const float4*)ue)[(size_t)n * 16 + q];
    else             v = ((const float4*)ie)[(size_t)(n - N_USERS) * 16 + q];
    float4* o = (float4*)(out + (size_t)n * ROWSTRIDE);
    float4 z = make_float4(0.f, 0.f, 0.f, 0.f);
    o[q]      = v;   // hop 0
    o[16 + q] = z;   // hop 1 accumulator
    o[32 + q] = z;   // hop 2 accumulator
    o[48 + q] = z;   // hop 3 accumulator
}

// ---------------- fused edge-dropout SpMM ----------------
// One wave cooperatively handles 32 edges at a time; lane l owns dims [2l,2l+1].
// Edge streams (rows/cols/vals/rand) are staged into per-wave LDS records of
// 16B/edge via gfx1250 async global->LDS loads, double buffered (ASYNCcnt
// pipeline). Each edge is then consumed with ONE ds_load_b128 broadcast.
__global__ void __launch_bounds__(256)
lgn_spmm_kernel(const int*   __restrict__ rows,
                const int*   __restrict__ cols,
                const float* __restrict__ vals,
                const float* __restrict__ erand,   // pre-offset to this hop
                float* __restrict__ out, int hop, int nnz) {
    // 16B per edge: [0]=row, [1]=col, [2]=val, [3]=rand  (8 KB total)
    __shared__ __align__(16) int s_edge[2][WPB][CHUNK][4];

    const int wave = threadIdx.x >> 5;
    const int lane = threadIdx.x & 31;

    const float* __restrict__ aggIn  = out + hop * DIM;        // gather slice
    float*       __restrict__ accOut = out + (hop + 1) * DIM;  // scatter slice

    const int nChunks    = (nnz + CHUNK - 1) / CHUNK;
    const int totalWaves = gridDim.x * WPB;
    int chunk = blockIdx.x * WPB + wave;
    if (chunk >= nChunks) return;

    // Issue 4 async b32 loads (one field of the edge record) into buf b.
    // Generic flat address of __shared__ truncated to 32 bits == wave-relative
    // LDS byte offset on gfx1250 (aperture tag lives in addr[63:32]).
    // NOTE: cannot fold the +4/+8/+12 into the instruction offset: the async
    // IOFFSET is added to BOTH the LDS and the global address.
    auto issue = [&](int c, int b) {
        int e = c * CHUNK + lane;
        if (e >= nnz) e = nnz - 1;                  // clamp; slot never consumed
        unsigned base = (unsigned)(unsigned long long)&s_edge[b][wave][lane][0];
        unsigned l1 = base + 4u, l2 = base + 8u, l3 = base + 12u;
        asm volatile("global_load_async_to_lds_b32 %0, %1, off"
                     :: "v"(base), "v"(rows  + e) : "memory");
        asm volatile("global_load_async_to_lds_b32 %0, %1, off"
                     :: "v"(l1),   "v"(cols  + e) : "memory");
        asm volatile("global_load_async_to_lds_b32 %0, %1, off"
                     :: "v"(l2),   "v"(vals  + e) : "memory");
        asm volatile("global_load_async_to_lds_b32 %0, %1, off"
                     :: "v"(l3),   "v"(erand + e) : "memory");
    };

    int buf = 0;
    issue(chunk, buf);                               // prime the pipeline
    for (;;) {
        const int  next    = chunk + totalWaves;
        const bool hasNext = next < nChunks;
        if (hasNext) {
            issue(next, buf ^ 1);                    // prefetch next chunk
            asm volatile("s_wait_asynccnt 0x4" ::: "memory"); // current 4 done
        } else {
            asm volatile("s_wait_asynccnt 0x0" ::: "memory");
        }

        int cnt = nnz - chunk * CHUNK;
        if (cnt > CHUNK) cnt = CHUNK;
        #pragma unroll 1
        for (int j = 0; j < cnt; ++j) {
            // one b128 LDS broadcast: whole edge record to all lanes
            const int4 e4 = *(const int4*)&s_edge[buf][wave][j][0];
            const float u = __int_as_float(e4.w);
            if (u >= 0.5f) {                         // floor(0.5 + u) == 1
                const int   r = e4.x;
                const int   c = e4.y;
                const float w = __int_as_float(e4.z) * 2.0f;  // 1/(1-0.5)
                const float2 x =
                    *(const float2*)(aggIn + (size_t)c * ROWSTRIDE + lane * 2);
                float* dst = accOut + (size_t)r * ROWSTRIDE + lane * 2;
                atomicAdd(dst,     w * x.x);
                atomicAdd(dst + 1, w * x.y);
            }
        }

        if (!hasNext) break;
        chunk = next;
        buf ^= 1;
    }
}

// ---------------- message dropout (in place on hop+1 slice) ----------------
__global__ void lgn_mdrop_kernel(const float* __restrict__ mrand,  // pre-offset
                                 float* __restrict__ out, int hop) {
    int idx = blockIdx.x * blockDim.x + threadIdx.x;
    const int total = N_NODES * (DIM / 4);
    if (idx >= total) return;
    int n = idx >> 4;
    int q = idx & 15;
    const float4 u = ((const float4*)mrand)[(size_t)n * 16 + q];
    float4* p = (float4*)(out + (size_t)n * ROWSTRIDE + (size_t)(hop + 1) * DIM);
    float4 v = p[q];
    const float s = 1.0f / 0.9f;                     // 1/(1-MESS_RATE)
    v.x = (u.x >= 0.1f) ? v.x * s : 0.0f;
    v.y = (u.y >= 0.1f) ? v.y * s : 0.0f;
    v.z = (u.z >= 0.1f) ? v.z * s : 0.0f;
    v.w = (u.w >= 0.1f) ? v.w * s : 0.0f;
    p[q] = v;
}

extern "C" void kernel_launch(void* const* d_in, const int* in_sizes, int n_in,
                              void* d_out, int out_size, void* d_ws, size_t ws_size,
                              hipStream_t stream) {
    (void)n_in; (void)out_size; (void)d_ws; (void)ws_size;
    const float* ue    = (const float*)d_in[0];   // [N_USERS, 64]
    const float* ie    = (const float*)d_in[1];   // [N_ITEMS, 64]
    const int*   rows  = (const int*)  d_in[2];   // [NNZ]
    const int*   cols  = (const int*)  d_in[3];   // [NNZ]
    const float* vals  = (const float*)d_in[4];   // [NNZ]
    const float* erand = (const float*)d_in[5];   // [3, NNZ]
    const float* mrand = (const float*)d_in[6];   // [3, N, 64]
    float* out = (float*)d_out;                   // [N, 4, 64] flat
    const int nnz = in_sizes[2];

    const int threads   = 256;
    const int ewBlocks  = (N_NODES * (DIM / 4) + threads - 1) / threads;
    const int spmBlocks = 3072;                   // 24576 waves over 250k chunks

    lgn_init_kernel<<<ewBlocks, threads, 0, stream>>>(ue, ie, out);
    for (int hop = 0; hop < NHOPS; ++hop) {
        lgn_spmm_kernel<<<spmBlocks, threads, 0, stream>>>(
            rows, cols, vals, erand + (size_t)hop * nnz, out, hop, nnz);
        lgn_mdrop_kernel<<<ewBlocks, threads, 0, stream>>>(
            mrand + (size_t)hop * N_NODES * DIM, out, hop);
    }
}